// CutOutput_82592221102778
// MI455X (gfx1250) — compile-verified
//
#include <hip/hip_runtime.h>
#include <hip/hip_bf16.h>

#define NUM_D    1571      // codes per visit
#define DPAD     2048      // next pow2 for bitonic sort
#define NVISITS  8192      // P*T = 128*64
#define BLOCK    256       // 8 wave32s

// Stable descending argsort per visit + prefix keep / zero-out.
// One workgroup per visit; bitonic sort of packed (orderable_value, index)
// 64-bit keys in LDS. Data staged into LDS via gfx1250 async-LDS B128 loads,
// with row misalignment absorbed by an LDS-side shift.
__global__ __launch_bounds__(BLOCK) void
CutOutput_82592221102778_kernel(const float* __restrict__ to_cut,
                                const int* __restrict__ base,
                                int* __restrict__ out)
{
    // 1576 = NUM_D rounded up to cover worst-case (shift=3) 16B-aligned superset.
    __shared__ __align__(16) float raw[1576];
    __shared__ unsigned long long keys[DPAD];

    const int n   = blockIdx.x;        // flat visit index (p*T + t)
    const int tid = threadIdx.x;

    // ---- Stage this visit's scores into LDS with async B128 loads (gfx1250).
    // Row base is only 4B-aligned (1571 floats/row); copy the 16B-aligned
    // superset and remember the float shift.
    const size_t row_start_f = (size_t)n * NUM_D;            // in floats
    const int    shift       = (int)(row_start_f & 3);       // 0..3 floats
    const float* abase       = to_cut + (row_start_f & ~(size_t)3); // 16B aligned
    const int    nchunk      = (NUM_D + shift + 3) >> 2;     // <= 394 x 16B

    for (int c = tid; c < nchunk; c += BLOCK) {
        unsigned           lds = (unsigned)(size_t)&raw[4 * c];        // addr[31:0] = LDS offset
        unsigned long long ga  = (unsigned long long)(size_t)(abase + 4 * c);
        asm volatile("global_load_async_to_lds_b128 %0, %1, off"
                     :: "v"(lds), "v"(ga) : "memory");
    }
#if __has_builtin(__builtin_amdgcn_s_wait_asynccnt)
    __builtin_amdgcn_s_wait_asynccnt(0);
#else
    asm volatile("s_wait_asynccnt 0" ::: "memory");
#endif
    __syncthreads();

    // ---- Build 64-bit sort keys: (desc-orderable float bits << 32) | index.
    // Ascending u64 sort == stable descending sort of the float values
    // (ties broken by ascending index, matching stable argsort(-x)).
    for (int i = tid; i < DPAD; i += BLOCK) {
        unsigned long long k;
        if (i < NUM_D) {
            unsigned u  = __float_as_uint(raw[shift + i]);
            unsigned dk = (u & 0x80000000u) ? u : (~u & 0x7FFFFFFFu);
            k = ((unsigned long long)dk << 32) | (unsigned)i;
        } else {
            k = ~0ULL;                 // padding sorts to the end
        }
        keys[i] = k;
    }
    __syncthreads();

    // ---- Bitonic sort (ascending) over DPAD u64 keys in LDS.
    for (int kk = 2; kk <= DPAD; kk <<= 1) {
        for (int j = kk >> 1; j > 0; j >>= 1) {
            for (int i = tid; i < DPAD; i += BLOCK) {
                int ixj = i ^ j;
                if (ixj > i) {
                    unsigned long long a = keys[i];
                    unsigned long long b = keys[ixj];
                    const bool up = ((i & kk) == 0);
                    if ((a > b) == up) { keys[i] = b; keys[ixj] = a; }
                }
            }
            __syncthreads();
        }
    }

    // ---- Emit: sorted index for positions < L, zero beyond.
    const int L = base[n];
    int* dst = out + (size_t)n * NUM_D;
    for (int i = tid; i < NUM_D; i += BLOCK) {
        int idx = (int)(unsigned)(keys[i] & 0xFFFFFFFFu);
        dst[i] = (i < L) ? idx : 0;
    }
}

extern "C" void kernel_launch(void* const* d_in, const int* in_sizes, int n_in,
                              void* d_out, int out_size, void* d_ws, size_t ws_size,
                              hipStream_t stream) {
    const float* to_cut = (const float*)d_in[0];   // [P, T, D] float32
    const int*   base   = (const int*)d_in[1];     // [1, N]    int32
    // d_in[2] (ids) is implied by the contiguous layout: visit n = p*T + t.
    int* out = (int*)d_out;                        // [P, T, D] int32

    CutOutput_82592221102778_kernel<<<NVISITS, BLOCK, 0, stream>>>(to_cut, base, out);
}